// ZCAWhitening_21517786153236
// MI455X (gfx1250) — compile-verified
//
#include <hip/hip_runtime.h>

#define EPSV 1e-5f
#define CCH  256
#define NB   64
#define HWSZ 3136
#define MTOT (NB * HWSZ)   // 200704

typedef __attribute__((ext_vector_type(2))) float v2f;
typedef __attribute__((ext_vector_type(8))) float v8f;

__device__ __forceinline__ v8f wmma4(v2f a, v2f b, v8f c) {
    // D = A(16x4) * B(4x16) + C(16x16), fp32
    return __builtin_amdgcn_wmma_f32_16x16x4_f32(false, a, false, b, (short)0, c, false, false);
}

// ---------------- per-channel mean ----------------
__global__ __launch_bounds__(256) void mean_kernel(const float* __restrict__ X,
                                                   float* __restrict__ mean) {
    int c = blockIdx.x;
    int t = threadIdx.x;
    float s = 0.f;
    for (int n = 0; n < NB; ++n) {
        const float* p = X + ((size_t)(n * CCH + c)) * HWSZ;
        for (int i = t; i < HWSZ; i += 256) s += p[i];
    }
    __shared__ float red[256];
    red[t] = s; __syncthreads();
    for (int off = 128; off > 0; off >>= 1) {
        if (t < off) red[t] += red[t + off];
        __syncthreads();
    }
    if (t == 0) mean[c] = red[0] / (float)MTOT;
}

__global__ void zero_kernel(float* __restrict__ p, int n) {
    int i = blockIdx.x * blockDim.x + threadIdx.x;
    if (i < n) p[i] = 0.f;
}

// ---------------- Gram = sum_m X[i,m] X[j,m] (uncentered) ----------------
// grid: (16 tile-blocks of 64x64, 64 M-slices); 256 threads = 8 waves
#define GSLICES 64
#define CHPS    49   // 3136 chunks / 64 slices

__global__ __launch_bounds__(256) void gram_kernel(const float* __restrict__ X,
                                                   float* __restrict__ gram) {
    const int tile  = blockIdx.x;            // 0..15
    const int slice = blockIdx.y;            // 0..63
    const int bi = (tile >> 2) * 64;
    const int bj = (tile & 3) * 64;
    const int t    = threadIdx.x;
    const int lane = t & 31;
    const int wave = t >> 5;                 // 0..7
    const int lr15 = lane & 15;
    const int lh   = lane >> 4;              // 0/1

    __shared__ float xa[64][68];
    __shared__ float xb[64][68];

    // each wave owns 2 of the 16 sub-tiles (4x4 grid of 16x16)
    const int t0 = wave * 2, t1 = wave * 2 + 1;
    const int i0a = (t0 >> 2) * 16, j0a = (t0 & 3) * 16;
    const int i0b = (t1 >> 2) * 16, j0b = (t1 & 3) * 16;
    v8f acc0 = {}; v8f acc1 = {};

    const int lrow = t >> 2;                 // 0..63 (loader row)
    const int lq   = t & 3;                  // loader quad

    for (int ch = 0; ch < CHPS; ++ch) {
        const int chunk = slice * CHPS + ch;     // 0..3135
        const int m0 = chunk * 64;
        const int n  = m0 / HWSZ;
        const int hw = m0 - n * HWSZ;            // multiple of 64
        const size_t basea = ((size_t)(n * CCH + bi + lrow)) * HWSZ + hw;
        const size_t baseb = ((size_t)(n * CCH + bj + lrow)) * HWSZ + hw;
        #pragma unroll
        for (int i = 0; i < 4; ++i) {
            const int c4 = (lq + 4 * i) * 4;
            float4 va = *(const float4*)(X + basea + c4);
            float4 vb = *(const float4*)(X + baseb + c4);
            xa[lrow][c4+0]=va.x; xa[lrow][c4+1]=va.y; xa[lrow][c4+2]=va.z; xa[lrow][c4+3]=va.w;
            xb[lrow][c4+0]=vb.x; xb[lrow][c4+1]=vb.y; xb[lrow][c4+2]=vb.z; xb[lrow][c4+3]=vb.w;
        }
        __syncthreads();
        #pragma unroll
        for (int km = 0; km < 64; km += 4) {
            const int colk = km + (lh << 1);
            v2f a0; a0.x = xa[i0a + lr15][colk]; a0.y = xa[i0a + lr15][colk + 1];
            v2f b0; b0.x = xb[j0a + lr15][colk]; b0.y = xb[j0a + lr15][colk + 1];
            acc0 = wmma4(a0, b0, acc0);
            v2f a1; a1.x = xa[i0b + lr15][colk]; a1.y = xa[i0b + lr15][colk + 1];
            v2f b1; b1.x = xb[j0b + lr15][colk]; b1.y = xb[j0b + lr15][colk + 1];
            acc1 = wmma4(a1, b1, acc1);
        }
        __syncthreads();
    }
    #pragma unroll
    for (int r = 0; r < 8; ++r) {
        const int row0 = bi + i0a + r + (lh << 3);
        const int col0 = bj + j0a + lr15;
        atomicAdd(&gram[row0 * CCH + col0], acc0[r]);
        const int row1 = bi + i0b + r + (lh << 3);
        const int col1 = bj + j0b + lr15;
        atomicAdd(&gram[row1 * CCH + col1], acc1[r]);
    }
}

// ---------------- cov = gram/M - mu mu^T + eps I ----------------
__global__ void cov_kernel(const float* __restrict__ gram, const float* __restrict__ mean,
                           float* __restrict__ cov) {
    int i = blockIdx.x, j = threadIdx.x;
    float v = gram[i * CCH + j] * (1.0f / (float)MTOT) - mean[i] * mean[j];
    if (i == j) v += EPSV;
    cov[i * CCH + j] = v;
}

__global__ __launch_bounds__(256) void trace_kernel(const float* __restrict__ cov,
                                                    float* __restrict__ s) {
    int t = threadIdx.x;
    __shared__ float red[256];
    red[t] = cov[t * CCH + t]; __syncthreads();
    for (int off = 128; off > 0; off >>= 1) {
        if (t < off) red[t] += red[t + off];
        __syncthreads();
    }
    if (t == 0) s[0] = red[0];
}

__global__ void ns_init_kernel(const float* __restrict__ cov, const float* __restrict__ s,
                               float* __restrict__ Y, float* __restrict__ Z) {
    int i = blockIdx.x, j = threadIdx.x;
    float inv = 1.0f / s[0];
    Y[i * CCH + j] = cov[i * CCH + j] * inv;
    Z[i * CCH + j] = (i == j) ? 1.0f : 0.0f;
}

// ---------------- generic 256x256x256 WMMA matmul ----------------
// mode 0: D = A@B ; mode 1: D = 1.5 I - 0.5 (A@B)
__global__ __launch_bounds__(128) void mm256_kernel(const float* __restrict__ A,
                                                    const float* __restrict__ B,
                                                    float* __restrict__ D, int mode) {
    const int id   = blockIdx.x * 4 + (threadIdx.x >> 5);  // 0..255 tile
    const int lane = threadIdx.x & 31;
    const int i0 = (id >> 4) << 4;
    const int j0 = (id & 15) << 4;
    const int lr = lane & 15;
    const int lh = lane >> 4;
    v8f acc = {};
    for (int k = 0; k < CCH; k += 4) {
        const int ka = k + (lh << 1);
        v2f a = *(const v2f*)(A + (i0 + lr) * CCH + ka);
        v2f b; b.x = B[ka * CCH + j0 + lr]; b.y = B[(ka + 1) * CCH + j0 + lr];
        acc = wmma4(a, b, acc);
    }
    #pragma unroll
    for (int r = 0; r < 8; ++r) {
        const int row = i0 + r + (lh << 3);
        const int col = j0 + lr;
        float v = acc[r];
        if (mode == 1) v = ((row == col) ? 1.5f : 0.0f) - 0.5f * v;
        D[row * CCH + col] = v;
    }
}

// ---------------- t[c] = (Z * rsqrt(s))[c,:] . mean ----------------
__global__ __launch_bounds__(256) void tvec_kernel(const float* __restrict__ Z,
                                                   const float* __restrict__ mean,
                                                   const float* __restrict__ s,
                                                   float* __restrict__ tvec) {
    int c = threadIdx.x;
    float rs = rsqrtf(s[0]);
    float acc = 0.f;
    for (int k = 0; k < CCH; ++k) acc += Z[c * CCH + k] * mean[k];
    tvec[c] = acc * rs;
}

// ---------------- whiten: out = W*(zca@X - t) + b, NCHW scatter ----------------
// grid: (M/128, C/64); 256 threads = 8 waves; each wave 4 of 32 16x16 tiles
__global__ __launch_bounds__(256) void whiten_kernel(const float* __restrict__ X,
                                                     const float* __restrict__ Z,
                                                     const float* __restrict__ sptr,
                                                     const float* __restrict__ tvec,
                                                     const float* __restrict__ weight,
                                                     const float* __restrict__ bias,
                                                     float* __restrict__ out) {
    const int cb = blockIdx.y * 64;
    const int mb = blockIdx.x * 128;
    const int t = threadIdx.x;
    const int lane = t & 31, wave = t >> 5;
    const int lr = lane & 15, lh = lane >> 4;
    const float rs = rsqrtf(sptr[0]);

    __shared__ float zs[64][36];    // 64 channels x 32 k
    __shared__ float xs[32][132];   // 32 k x 128 m

    v8f acc[4] = {{}, {}, {}, {}};

    for (int kb = 0; kb < CCH; kb += 32) {
        // stage zca chunk (pre-scaled by rsqrt(trace))
        #pragma unroll
        for (int i = 0; i < 2; ++i) {
            const int f = t + i * 256;        // 0..511
            const int r = f >> 3;             // 0..63
            const int q = (f & 7) * 4;        // 0..28 step 4
            float4 v = *(const float4*)(Z + (cb + r) * CCH + kb + q);
            zs[r][q+0] = v.x * rs; zs[r][q+1] = v.y * rs;
            zs[r][q+2] = v.z * rs; zs[r][q+3] = v.w * rs;
        }
        // stage X chunk (k rows = channels kb..kb+31, m cols mb..mb+127)
        #pragma unroll
        for (int i = 0; i < 4; ++i) {
            const int f = t + i * 256;        // 0..1023
            const int r = f >> 5;             // 0..31
            const int q = (f & 31) * 4;       // 0..124 step 4
            const int m = mb + q;
            const int n = m / HWSZ;
            const int hw = m - n * HWSZ;      // 4-aligned within one n
            float4 v = *(const float4*)(X + ((size_t)(n * CCH + kb + r)) * HWSZ + hw);
            xs[r][q+0] = v.x; xs[r][q+1] = v.y; xs[r][q+2] = v.z; xs[r][q+3] = v.w;
        }
        __syncthreads();
        #pragma unroll
        for (int p = 0; p < 4; ++p) {
            const int id = wave * 4 + p;      // 0..31
            const int c0 = (id >> 3) << 4;
            const int m0 = (id & 7) << 4;
            #pragma unroll
            for (int km = 0; km < 32; km += 4) {
                const int ca = km + (lh << 1);
                v2f a; a.x = zs[c0 + lr][ca]; a.y = zs[c0 + lr][ca + 1];
                v2f b; b.x = xs[ca][m0 + lr]; b.y = xs[ca + 1][m0 + lr];
                acc[p] = wmma4(a, b, acc[p]);
            }
        }
        __syncthreads();
    }
    #pragma unroll
    for (int p = 0; p < 4; ++p) {
        const int id = wave * 4 + p;
        const int c0 = (id >> 3) << 4;
        const int m0 = (id & 7) << 4;
        const int m = mb + m0 + lr;
        const int n = m / HWSZ;
        const int hw = m - n * HWSZ;
        #pragma unroll
        for (int r = 0; r < 8; ++r) {
            const int c = cb + c0 + r + (lh << 3);
            const float v = weight[c] * (acc[p][r] - tvec[c]) + bias[c];
            out[((size_t)(n * CCH + c)) * HWSZ + hw] = v;
        }
    }
}

extern "C" void kernel_launch(void* const* d_in, const int* in_sizes, int n_in,
                              void* d_out, int out_size, void* d_ws, size_t ws_size,
                              hipStream_t stream) {
    (void)in_sizes; (void)n_in; (void)out_size; (void)ws_size;
    const float* X      = (const float*)d_in[0];
    const float* weight = (const float*)d_in[1];
    const float* bias   = (const float*)d_in[2];
    float* out = (float*)d_out;
    float* ws  = (float*)d_ws;

    float* mean = ws;                 // 256
    float* tvec = ws + 256;           // 256
    float* sval = ws + 512;           // 1 (padded)
    float* gram = ws + 768;           // 65536
    float* cov  = gram + 65536;
    float* Y    = cov  + 65536;
    float* Z    = Y    + 65536;
    float* T    = Z    + 65536;
    float* Y2   = T    + 65536;
    float* Z2   = Y2   + 65536;       // total ~1.84 MB

    mean_kernel<<<CCH, 256, 0, stream>>>(X, mean);
    zero_kernel<<<(CCH * CCH + 255) / 256, 256, 0, stream>>>(gram, CCH * CCH);
    gram_kernel<<<dim3(16, GSLICES), 256, 0, stream>>>(X, gram);
    cov_kernel<<<CCH, CCH, 0, stream>>>(gram, mean, cov);
    trace_kernel<<<1, 256, 0, stream>>>(cov, sval);
    ns_init_kernel<<<CCH, CCH, 0, stream>>>(cov, sval, Y, Z);

    float *y = Y, *z = Z, *y2 = Y2, *z2 = Z2;
    for (int it = 0; it < 20; ++it) {
        mm256_kernel<<<64, 128, 0, stream>>>(z, y, T, 1);   // T = 1.5I - 0.5 Z Y
        mm256_kernel<<<64, 128, 0, stream>>>(y, T, y2, 0);  // Y' = Y T
        mm256_kernel<<<64, 128, 0, stream>>>(T, z, z2, 0);  // Z' = T Z
        float* tmp;
        tmp = y; y = y2; y2 = tmp;
        tmp = z; z = z2; z2 = tmp;
    }
    tvec_kernel<<<1, 256, 0, stream>>>(z, mean, sval, tvec);
    whiten_kernel<<<dim3(MTOT / 128, CCH / 64), 256, 0, stream>>>(X, z, sval, tvec,
                                                                  weight, bias, out);
}